// PDE_47072841564544
// MI455X (gfx1250) — compile-verified
//
#include <hip/hip_runtime.h>

// Reaction-diffusion RHS:  zdot = Basis(z)·P  +  diag(d)·Laplacian(z)
// z: [8, 2, 1024, 1024] fp32.  Memory-bound: 128 MiB @ 23.3 TB/s ~ 5.6 us floor.
// Reaction GEMM (pixels x 10) @ (10 x 2) mapped onto V_WMMA_F32_16X16X4_F32.

typedef __attribute__((ext_vector_type(2))) float v2f;
typedef __attribute__((ext_vector_type(8))) float v8f;

#define HH 1024
#define WW 1024
#define PLANE (HH * WW)       // 1 << 20
#define BSTRIDE (2 * PLANE)   // per-batch stride (2 channels)

__global__ __launch_bounds__(256) void rd_rhs_kernel(
    const float* __restrict__ z,     // [8][2][1024][1024]
    const float* __restrict__ rp,    // [10][2] reaction params (row-major l*2+c)
    const float* __restrict__ dp,    // [2] diffusion params
    float* __restrict__ out)         // [8][2][1024][1024]
{
    const int tid  = blockIdx.x * 256 + threadIdx.x;
    const int lane = threadIdx.x & 31;
    const int wave = threadIdx.x >> 5;
    const int half = lane >> 4;      // which 16-lane half of the wave
    const int q    = lane & 15;      // column index inside the half

    // ---- pixel coordinates (one pixel per lane, exact grid coverage) ----
    const int w = tid & (WW - 1);
    const int h = (tid >> 10) & (HH - 1);
    const int b = tid >> 20;

    const float* z0 = z + (size_t)b * BSTRIDE;      // channel-0 plane
    const float* z1 = z0 + PLANE;                   // channel-1 plane

    const int row  = h * WW;
    const int rowm = ((h - 1) & (HH - 1)) * WW;
    const int rowp = ((h + 1) & (HH - 1)) * WW;
    const int wm   = (w - 1) & (WW - 1);
    const int wp   = (w + 1) & (WW - 1);

    const float u = z0[row + w];
    const float v = z1[row + w];
    const float lap_u = z0[rowm + w] + z0[rowp + w] + z0[row + wm] + z0[row + wp] - 4.0f * u;
    const float lap_v = z1[rowm + w] + z1[rowp + w] + z1[row + wm] + z1[row + wp] - 4.0f * v;

    const float d0 = dp[0];
    const float d1 = dp[1];

    // ---- 10 monomials of total degree <= 3, zero-padded to K=12 ----
    float bb[12];
    bb[0] = 1.0f; bb[1] = u; bb[2] = v;
    bb[3] = u * u; bb[4] = u * v; bb[5] = v * v;
    bb[6] = bb[3] * u; bb[7] = bb[3] * v; bb[8] = u * bb[5]; bb[9] = bb[5] * v;
    bb[10] = 0.0f; bb[11] = 0.0f;

    // Partner pixel's basis (lane ^ 16) for the K=2/3 slots of the A layout.
    float ob[12];
    #pragma unroll
    for (int i = 0; i < 10; ++i) ob[i] = __shfl_xor(bb[i], 16, 32);
    ob[10] = 0.0f; ob[11] = 0.0f;

    // ---- B matrix: 12x16, rows = basis index, cols 0/1 = reaction params ----
    // VGPR0: rows k (lanes 0-15) / k+2 (lanes 16-31); VGPR1: rows k+1 / k+3.
    v2f Bm[3];
    #pragma unroll
    for (int c = 0; c < 3; ++c) {
        const int k  = 4 * c;
        const int r0 = k + 2 * half;
        const int r1 = k + 1 + 2 * half;
        Bm[c].x = (q < 2 && r0 < 10) ? rp[r0 * 2 + q] : 0.0f;
        Bm[c].y = (q < 2 && r1 < 10) ? rp[r1 * 2 + q] : 0.0f;
    }

    // ---- WMMA: acc1 covers pixels 0-15 of the wave, acc2 covers 16-31 ----
    v8f acc1 = {0.f, 0.f, 0.f, 0.f, 0.f, 0.f, 0.f, 0.f};
    v8f acc2 = {0.f, 0.f, 0.f, 0.f, 0.f, 0.f, 0.f, 0.f};
    #pragma unroll
    for (int c = 0; c < 3; ++c) {
        const int k = 4 * c;
        v2f a1, a2;
        // A layout (16x4 f32): lanes 0-15 -> (M=lane, K=k/k+1),
        //                      lanes 16-31 -> (M=lane-16, K=k+2/k+3)
        a1.x = half ? ob[k + 2] : bb[k];
        a1.y = half ? ob[k + 3] : bb[k + 1];
        a2.x = half ? bb[k + 2] : ob[k];
        a2.y = half ? bb[k + 3] : ob[k + 1];
        acc1 = __builtin_amdgcn_wmma_f32_16x16x4_f32(
            false, a1, false, Bm[c], (short)0, acc1, false, false);
        acc2 = __builtin_amdgcn_wmma_f32_16x16x4_f32(
            false, a2, false, Bm[c], (short)0, acc2, false, false);
    }

    // ---- redistribute D-matrix (only N=0,1 columns are live) via LDS ----
    // D layout: VGPR r, lanes 0-15 -> (M=r, N=lane); lanes 16-31 -> (M=8+r, N=lane-16)
    __shared__ float lds[8][2][32];
    if (q < 2) {
        #pragma unroll
        for (int r = 0; r < 8; ++r) {
            lds[wave][q][half * 8 + r]      = acc1[r];  // pixels 0..15
            lds[wave][q][16 + half * 8 + r] = acc2[r];  // pixels 16..31
        }
    }
    // Same-wave LDS ops are serviced in order; explicit split-counter wait for safety.
    asm volatile("s_wait_dscnt 0x0" ::: "memory");
    const float R0 = lds[wave][0][lane];
    const float R1 = lds[wave][1][lane];

    // ---- combine and store (fully coalesced 128B per plane per wave) ----
    float* o0 = out + (size_t)b * BSTRIDE + row + w;
    o0[0]     = R0 + d0 * lap_u;
    o0[PLANE] = R1 + d1 * lap_v;
}

extern "C" void kernel_launch(void* const* d_in, const int* in_sizes, int n_in,
                              void* d_out, int out_size, void* d_ws, size_t ws_size,
                              hipStream_t stream) {
    const float* z  = (const float*)d_in[0];   // [8,2,1024,1024]
    const float* rp = (const float*)d_in[1];   // [10,2]
    const float* dp = (const float*)d_in[2];   // [2]
    float* out = (float*)d_out;

    // 8*1024*1024 pixels, one per thread: 32768 blocks x 256 threads (no partial waves)
    dim3 grid(32768), block(256);
    hipLaunchKernelGGL(rd_rhs_kernel, grid, block, 0, stream, z, rp, dp, out);
}